// GraphSAGEEncoder_81544249081903
// MI455X (gfx1250) — compile-verified
//
#include <hip/hip_runtime.h>

typedef __attribute__((ext_vector_type(2))) float v2f;
typedef __attribute__((ext_vector_type(8))) float v8f;

#define NNODES 100000
#define NEDGES 600000

__device__ __forceinline__ v8f wmma4(v2f a, v2f b, v8f c) {
  return __builtin_amdgcn_wmma_f32_16x16x4_f32(false, a, false, b, (short)0, c,
                                               false, false);
}

// ---------------------------------------------------------------- zero scratch
__global__ void sage_zero_kernel(float4* __restrict__ p, int n4) {
  int i = blockIdx.x * blockDim.x + threadIdx.x;
  if (i < n4) p[i] = make_float4(0.0f, 0.0f, 0.0f, 0.0f);
}

// ------------------------------------------- layer-1 scatter (F=128) + degree
__global__ void sage_scatter1_kernel(const float* __restrict__ x,
                                     const int* __restrict__ ei,
                                     float* __restrict__ agg,
                                     float* __restrict__ cnt) {
  int t = blockIdx.x * blockDim.x + threadIdx.x;
  int e = t >> 5;
  int lane = t & 31;
  if (e >= NEDGES) return;
  int src = ei[e];
  int dst = ei[NEDGES + e];
  float4 v = ((const float4*)(x + (size_t)src * 128))[lane];  // 32 lanes * 4 = 128
  float* a = agg + (size_t)dst * 128 + lane * 4;
  atomicAdd(a + 0, v.x);
  atomicAdd(a + 1, v.y);
  atomicAdd(a + 2, v.z);
  atomicAdd(a + 3, v.w);
  if (lane == 0) atomicAdd(cnt + dst, 1.0f);
}

// ------------------------------------------- layer-2 scatter (F=64)
__global__ void sage_scatter2_kernel(const float* __restrict__ h,
                                     const int* __restrict__ ei,
                                     float* __restrict__ agg) {
  int t = blockIdx.x * blockDim.x + threadIdx.x;
  int e = t >> 5;
  int lane = t & 31;
  if (e >= NEDGES) return;
  int src = ei[e];
  int dst = ei[NEDGES + e];
  float2 v = ((const float2*)(h + (size_t)src * 64))[lane];   // 32 lanes * 2 = 64
  float* a = agg + (size_t)dst * 64 + lane * 2;
  atomicAdd(a + 0, v.x);
  atomicAdd(a + 1, v.y);
}

// --------------------------------- layer 1: h = relu([agg/cnt | x] @ W1^T + b1)
// One wave per 32 nodes (2 M-tiles); 4 N-tiles (64 outputs).
// 100000 % 32 == 0 -> no tail, no predication; EXEC all-ones throughout.
__global__ void sage_gemm1_kernel(const float* __restrict__ x,
                                  const float* __restrict__ agg,
                                  const float* __restrict__ cnt,
                                  const float* __restrict__ wl,   // [64,128]
                                  const float* __restrict__ bl,   // [64]
                                  const float* __restrict__ wr,   // [64,128]
                                  float* __restrict__ h) {        // [N,64]
  int wave = (blockIdx.x * blockDim.x + threadIdx.x) >> 5;
  int lane = threadIdx.x & 31;
  if (wave >= NNODES / 32) return;          // uniform per wave
  int node0 = wave * 32;
  int row   = lane & 15;                    // M for A-frag, N for B-frag
  int khalf = lane >> 4;                    // lanes 0-15: K=k0,k0+1 ; 16-31: +2,+3

  int   roff[2];
  float rinv[2];
#pragma unroll
  for (int mt = 0; mt < 2; ++mt) {
    int node = node0 + mt * 16 + row;
    roff[mt] = node * 128;
    rinv[mt] = 1.0f / fmaxf(cnt[node], 1.0f);
  }

  v8f acc[2][4] = {};

  // ---- K in [0,128): mean(neighbors) x w1_l
  for (int k0 = 0; k0 < 128; k0 += 4) {
    int ka = k0 + khalf * 2;
    v2f a[2], b[4];
#pragma unroll
    for (int mt = 0; mt < 2; ++mt) {
      float2 t = *(const float2*)(agg + roff[mt] + ka);
      a[mt][0] = t.x * rinv[mt];
      a[mt][1] = t.y * rinv[mt];
    }
    const float* wbase = wl + row * 128 + ka;       // w1_l[nt*16+row][ka..ka+1]
#pragma unroll
    for (int nt = 0; nt < 4; ++nt) {
      float2 t = *(const float2*)(wbase + nt * 2048);
      b[nt][0] = t.x; b[nt][1] = t.y;
    }
#pragma unroll
    for (int mt = 0; mt < 2; ++mt)
#pragma unroll
      for (int nt = 0; nt < 4; ++nt)
        acc[mt][nt] = wmma4(a[mt], b[nt], acc[mt][nt]);
  }

  // ---- K in [128,256): self features x w1_r
  for (int k0 = 0; k0 < 128; k0 += 4) {
    int ka = k0 + khalf * 2;
    v2f a[2], b[4];
#pragma unroll
    for (int mt = 0; mt < 2; ++mt) {
      float2 t = *(const float2*)(x + roff[mt] + ka);
      a[mt][0] = t.x; a[mt][1] = t.y;
    }
    const float* wbase = wr + row * 128 + ka;
#pragma unroll
    for (int nt = 0; nt < 4; ++nt) {
      float2 t = *(const float2*)(wbase + nt * 2048);
      b[nt][0] = t.x; b[nt][1] = t.y;
    }
#pragma unroll
    for (int mt = 0; mt < 2; ++mt)
#pragma unroll
      for (int nt = 0; nt < 4; ++nt)
        acc[mt][nt] = wmma4(a[mt], b[nt], acc[mt][nt]);
  }

  // C/D layout: VGPR i -> M = i + 8*khalf, N = row
  int mbase = khalf * 8;
#pragma unroll
  for (int mt = 0; mt < 2; ++mt) {
#pragma unroll
    for (int nt = 0; nt < 4; ++nt) {
      float bias = bl[nt * 16 + row];
#pragma unroll
      for (int i = 0; i < 8; ++i) {
        int nodeS = node0 + mt * 16 + mbase + i;
        h[(size_t)nodeS * 64 + nt * 16 + row] =
            fmaxf(acc[mt][nt][i] + bias, 0.0f);
      }
    }
  }
}

// --------------------------------- layer 2: out = [agg2/cnt | h] @ W2^T + b2
__global__ void sage_gemm2_kernel(const float* __restrict__ h,
                                  const float* __restrict__ agg,
                                  const float* __restrict__ cnt,
                                  const float* __restrict__ wl,   // [32,64]
                                  const float* __restrict__ bl,   // [32]
                                  const float* __restrict__ wr,   // [32,64]
                                  float* __restrict__ out) {      // [N,32]
  int wave = (blockIdx.x * blockDim.x + threadIdx.x) >> 5;
  int lane = threadIdx.x & 31;
  if (wave >= NNODES / 32) return;
  int node0 = wave * 32;
  int row   = lane & 15;
  int khalf = lane >> 4;

  int   roff[2];
  float rinv[2];
#pragma unroll
  for (int mt = 0; mt < 2; ++mt) {
    int node = node0 + mt * 16 + row;
    roff[mt] = node * 64;
    rinv[mt] = 1.0f / fmaxf(cnt[node], 1.0f);
  }

  v8f acc[2][2] = {};

  // ---- K in [0,64): mean(neighbors) x w2_l
  for (int k0 = 0; k0 < 64; k0 += 4) {
    int ka = k0 + khalf * 2;
    v2f a[2], b[2];
#pragma unroll
    for (int mt = 0; mt < 2; ++mt) {
      float2 t = *(const float2*)(agg + roff[mt] + ka);
      a[mt][0] = t.x * rinv[mt];
      a[mt][1] = t.y * rinv[mt];
    }
    const float* wbase = wl + row * 64 + ka;        // [32,64]: tile stride 1024
#pragma unroll
    for (int nt = 0; nt < 2; ++nt) {
      float2 t = *(const float2*)(wbase + nt * 1024);
      b[nt][0] = t.x; b[nt][1] = t.y;
    }
#pragma unroll
    for (int mt = 0; mt < 2; ++mt)
#pragma unroll
      for (int nt = 0; nt < 2; ++nt)
        acc[mt][nt] = wmma4(a[mt], b[nt], acc[mt][nt]);
  }

  // ---- K in [64,128): hidden features x w2_r
  for (int k0 = 0; k0 < 64; k0 += 4) {
    int ka = k0 + khalf * 2;
    v2f a[2], b[2];
#pragma unroll
    for (int mt = 0; mt < 2; ++mt) {
      float2 t = *(const float2*)(h + roff[mt] + ka);
      a[mt][0] = t.x; a[mt][1] = t.y;
    }
    const float* wbase = wr + row * 64 + ka;
#pragma unroll
    for (int nt = 0; nt < 2; ++nt) {
      float2 t = *(const float2*)(wbase + nt * 1024);
      b[nt][0] = t.x; b[nt][1] = t.y;
    }
#pragma unroll
    for (int mt = 0; mt < 2; ++mt)
#pragma unroll
      for (int nt = 0; nt < 2; ++nt)
        acc[mt][nt] = wmma4(a[mt], b[nt], acc[mt][nt]);
  }

  int mbase = khalf * 8;
#pragma unroll
  for (int mt = 0; mt < 2; ++mt) {
#pragma unroll
    for (int nt = 0; nt < 2; ++nt) {
      float bias = bl[nt * 16 + row];
#pragma unroll
      for (int i = 0; i < 8; ++i) {
        int nodeS = node0 + mt * 16 + mbase + i;
        out[(size_t)nodeS * 32 + nt * 16 + row] = acc[mt][nt][i] + bias;
      }
    }
  }
}

// ---------------------------------------------------------------- launch
extern "C" void kernel_launch(void* const* d_in, const int* in_sizes, int n_in,
                              void* d_out, int out_size, void* d_ws, size_t ws_size,
                              hipStream_t stream) {
  (void)in_sizes; (void)n_in; (void)out_size; (void)ws_size;

  const float* x   = (const float*)d_in[0];
  const int*   ei  = (const int*)d_in[1];      // edge_index [2, E] (int32)
  const float* w1l = (const float*)d_in[2];
  const float* b1l = (const float*)d_in[3];
  const float* w1r = (const float*)d_in[4];
  const float* w2l = (const float*)d_in[5];
  const float* b2l = (const float*)d_in[6];
  const float* w2r = (const float*)d_in[7];
  float* out = (float*)d_out;

  // workspace layout (floats): agg1[N*128] | agg2[N*64] | cnt[N] | h[N*64]
  float* agg1 = (float*)d_ws;
  float* agg2 = agg1 + (size_t)NNODES * 128;
  float* cntb = agg2 + (size_t)NNODES * 64;
  float* hbuf = cntb + (size_t)NNODES;

  // zero agg1+agg2+cnt (contiguous prefix): 19,300,000 floats = 4,825,000 float4
  int n4 = (NNODES * 128 + NNODES * 64 + NNODES) / 4;
  sage_zero_kernel<<<(n4 + 255) / 256, 256, 0, stream>>>((float4*)d_ws, n4);

  int sc_threads = NEDGES * 32;
  sage_scatter1_kernel<<<(sc_threads + 255) / 256, 256, 0, stream>>>(x, ei, agg1, cntb);

  int gemm_waves = NNODES / 32;             // one wave32 per 32-node block (no tail)
  int gemm_threads = gemm_waves * 32;
  sage_gemm1_kernel<<<(gemm_threads + 255) / 256, 256, 0, stream>>>(
      x, agg1, cntb, w1l, b1l, w1r, hbuf);

  sage_scatter2_kernel<<<(sc_threads + 255) / 256, 256, 0, stream>>>(hbuf, ei, agg2);

  sage_gemm2_kernel<<<(gemm_threads + 255) / 256, 256, 0, stream>>>(
      hbuf, agg2, cntb, w2l, b2l, w2r, out);
}